// SelfAttention_61220463837579
// MI455X (gfx1250) — compile-verified
//
#include <hip/hip_runtime.h>

// ---------------------------------------------------------------------------
// Problem constants (from reference setup_inputs)
// ---------------------------------------------------------------------------
#define HID   256
#define LK    2048
#define LQ    64
#define BSZ   8
#define MCAND 16
#define NKEY  (BSZ * MCAND)        // 128 key blocks
#define SPLIT 8                    // split-K for the big column-sum reduction
#define ROWS_PER_SPLIT (LK / SPLIT) // 256 rows per partial block

typedef __attribute__((ext_vector_type(2))) float v2f;
typedef __attribute__((ext_vector_type(8))) float v8f;

// ---------------------------------------------------------------------------
// Kernel 1: partial column sums of key[n, k, h] over k-chunks.
// Dominant kernel: streams 256 MB of key. float4 coalesced, deterministic.
// grid = NKEY*SPLIT blocks, 256 threads.
// ---------------------------------------------------------------------------
__global__ __launch_bounds__(256) void key_colsum_partial(
    const float* __restrict__ key, float* __restrict__ partial) {
  __shared__ float4 red[256];
  const int blk   = blockIdx.x;
  const int n     = blk >> 3;          // key block index 0..127
  const int s     = blk & (SPLIT - 1); // k-chunk
  const int tid   = threadIdx.x;
  const int c64   = tid & 63;          // which float4 column group (0..63)
  const int phase = tid >> 6;          // row phase 0..3

  const float4* k4 = (const float4*)key + (size_t)n * (LK * HID / 4);

  float4 acc; acc.x = acc.y = acc.z = acc.w = 0.f;
  int kk = s * ROWS_PER_SPLIT + phase;
  #pragma unroll 4
  for (int i = 0; i < ROWS_PER_SPLIT / 4; ++i, kk += 4) {
    float4 v = k4[(size_t)kk * (HID / 4) + c64];
    acc.x += v.x; acc.y += v.y; acc.z += v.z; acc.w += v.w;
  }
  red[tid] = acc;
  __syncthreads();
  if (tid < 64) {
    float4 a = red[tid], b = red[tid + 64], c = red[tid + 128], d = red[tid + 192];
    float4 r;
    r.x = a.x + b.x + c.x + d.x;
    r.y = a.y + b.y + c.y + d.y;
    r.z = a.z + b.z + c.z + d.z;
    r.w = a.w + b.w + c.w + d.w;
    ((float4*)partial)[(size_t)(n * SPLIT + s) * (HID / 4) + tid] = r;
  }
}

// ---------------------------------------------------------------------------
// Kernel 2: combine SPLIT partials -> ksum[n][h]. grid = NKEY, 256 threads.
// ---------------------------------------------------------------------------
__global__ __launch_bounds__(256) void key_colsum_final(
    const float* __restrict__ partial, float* __restrict__ ksum) {
  const int n = blockIdx.x;
  const int h = threadIdx.x;
  float s = 0.f;
  #pragma unroll
  for (int i = 0; i < SPLIT; ++i)
    s += partial[(size_t)(n * SPLIT + i) * HID + h];
  ksum[(size_t)n * HID + h] = s;
}

// ---------------------------------------------------------------------------
// Kernel 3: single-block attention pipeline with WMMA f32 16x16x4 GEMMs.
//   qs[b,h]   = sum_q query[b,q,h]                        (pad M: 8 -> 16)
//   qsum      = qs @ w_q^T + 64*b_q        (WMMA GEMM 1, B[k][n] = w_q[n*H+k])
//   tmat      = qsum @ w_k                 (WMMA GEMM 2, B[k][n] = w_k[k*H+n])
//   att[b,m]  = (tmat[b]·ksum[b*16+m]/LK + qsum[b]·b_k) / 16
//   idx[b]    = argmax_m att[b,m]   (strict '>' keeps first max, like argmax)
// 256 threads = 8 waves; each wave owns 2 N-tiles of 16.
// ---------------------------------------------------------------------------
__global__ __launch_bounds__(256) void attention_kernel(
    const float* __restrict__ query,
    const float* __restrict__ wq, const float* __restrict__ bq,
    const float* __restrict__ wk, const float* __restrict__ bk,
    const float* __restrict__ ksum, int* __restrict__ idx_out) {
  __shared__ float qs  [16][HID];
  __shared__ float qsum[16][HID];
  __shared__ float tmat[16][HID];
  __shared__ float att [NKEY];
  __shared__ float qb  [BSZ];

  const int tid    = threadIdx.x;
  const int wave   = tid >> 5;
  const int lane   = tid & 31;
  const int laneLo = lane & 15;   // matrix row (A/M) or column (B,N)
  const int hi     = lane >> 4;   // K half selector for 16x16x4 layout

  // ---- stage A: qs = sum over Lq of query (rows 8..15 zero-padded) ----
  for (int b = 0; b < BSZ; ++b) {
    const float* qp = query + ((size_t)b * LQ) * HID + tid;
    float s = 0.f;
    #pragma unroll 8
    for (int q = 0; q < LQ; ++q) s += qp[(size_t)q * HID];
    qs[b][tid]     = s;
    qs[b + 8][tid] = 0.f;
  }
  __syncthreads();

  // ---- stage B: GEMM1  qsum = qs @ w_q^T + 64*b_q  (K = HID) ----
  for (int tI = 0; tI < 2; ++tI) {
    const int n0 = wave * 32 + tI * 16;
    v8f c = {0.f, 0.f, 0.f, 0.f, 0.f, 0.f, 0.f, 0.f};
    for (int kk = 0; kk < HID; kk += 4) {
      const int ka = kk + 2 * hi;
      v2f a, b;
      a.x = qs[laneLo][ka];
      a.y = qs[laneLo][ka + 1];
      const float* wrow = wq + (size_t)(n0 + laneLo) * HID + ka;
      b.x = wrow[0];
      b.y = wrow[1];
      c = __builtin_amdgcn_wmma_f32_16x16x4_f32(
          false, a, false, b, (short)0, c, false, false);
    }
    const int col  = n0 + laneLo;
    const float bias = 64.f * bq[col];
    #pragma unroll
    for (int r = 0; r < 8; ++r) qsum[r + 8 * hi][col] = c[r] + bias;
  }
  __syncthreads();

  // ---- stage C: GEMM2  tmat = qsum @ w_k  (K = HID) ----
  for (int tI = 0; tI < 2; ++tI) {
    const int n0 = wave * 32 + tI * 16;
    v8f c = {0.f, 0.f, 0.f, 0.f, 0.f, 0.f, 0.f, 0.f};
    for (int kk = 0; kk < HID; kk += 4) {
      const int ka = kk + 2 * hi;
      v2f a, b;
      a.x = qsum[laneLo][ka];
      a.y = qsum[laneLo][ka + 1];
      b.x = wk[(size_t)ka * HID + n0 + laneLo];
      b.y = wk[(size_t)(ka + 1) * HID + n0 + laneLo];
      c = __builtin_amdgcn_wmma_f32_16x16x4_f32(
          false, a, false, b, (short)0, c, false, false);
    }
    const int col = n0 + laneLo;
    #pragma unroll
    for (int r = 0; r < 8; ++r) tmat[r + 8 * hi][col] = c[r];
  }
  __syncthreads();

  // ---- stage D: bias dot  qb[b] = qsum[b]·b_k ----
  if (tid < BSZ) {
    float s = 0.f;
    for (int h = 0; h < HID; ++h) s += qsum[tid][h] * bk[h];
    qb[tid] = s;
  }
  __syncthreads();

  // ---- stage E: attention scores (one thread per (b,m) pair) ----
  if (tid < NKEY) {
    const int b = tid >> 4;
    const float* kr = ksum + (size_t)tid * HID;
    float acc = 0.f;
    for (int h = 0; h < HID; ++h) acc += tmat[b][h] * kr[h];
    att[tid] = (acc * (1.0f / (float)LK) + qb[b]) * (1.0f / 16.0f);
  }
  __syncthreads();

  // ---- stage F: per-batch argmax (first-max wins, matching jnp.argmax) ----
  if (tid < BSZ) {
    int best = 0;
    float bv = att[tid * MCAND];
    for (int m = 1; m < MCAND; ++m) {
      float v = att[tid * MCAND + m];
      if (v > bv) { bv = v; best = m; }
    }
    idx_out[tid] = best;
  }
}

// ---------------------------------------------------------------------------
// Kernel 4: gather winning raw key block per batch (bitwise-exact copy).
// ---------------------------------------------------------------------------
__global__ __launch_bounds__(256) void gather_kernel(
    const float* __restrict__ key, const int* __restrict__ idx,
    float* __restrict__ out) {
  const size_t f4_per_b = (size_t)LK * HID / 4;   // 131072
  const size_t total    = (size_t)BSZ * f4_per_b; // 1,048,576 float4
  const float4* k4 = (const float4*)key;
  float4* o4 = (float4*)out;
  for (size_t g = (size_t)blockIdx.x * blockDim.x + threadIdx.x; g < total;
       g += (size_t)gridDim.x * blockDim.x) {
    const size_t b   = g / f4_per_b;
    const size_t off = g - b * f4_per_b;
    const int id = idx[b];
    o4[g] = k4[((size_t)(b * MCAND + id)) * f4_per_b + off];
  }
}

// ---------------------------------------------------------------------------
// Launcher
// ---------------------------------------------------------------------------
extern "C" void kernel_launch(void* const* d_in, const int* in_sizes, int n_in,
                              void* d_out, int out_size, void* d_ws, size_t ws_size,
                              hipStream_t stream) {
  const float* query = (const float*)d_in[0];
  const float* key   = (const float*)d_in[1];
  const float* wq    = (const float*)d_in[2];
  const float* bq    = (const float*)d_in[3];
  const float* wk    = (const float*)d_in[4];
  const float* bk    = (const float*)d_in[5];
  float* out = (float*)d_out;

  // workspace layout (floats): partial[128*8*256] | ksum[128*256] | idx[8]
  float* partial = (float*)d_ws;
  float* ksum    = partial + (size_t)NKEY * SPLIT * HID;
  int*   idx     = (int*)(ksum + (size_t)NKEY * HID);

  key_colsum_partial<<<NKEY * SPLIT, 256, 0, stream>>>(key, partial);
  key_colsum_final  <<<NKEY,         256, 0, stream>>>(partial, ksum);
  attention_kernel  <<<1,            256, 0, stream>>>(query, wq, bq, wk, bk, ksum, idx);
  gather_kernel     <<<2048,         256, 0, stream>>>(key, idx, out);
}